// ELSA_46488726011980
// MI455X (gfx1250) — compile-verified
//
#include <hip/hip_runtime.h>
#include <math.h>

// Problem constants (from reference)
#define BB    4
#define HH    56
#define WW    56
#define DIMC  256
#define GG    8
#define KSZ   5
#define DQK   176
#define DV    256
#define NPIX  (BB*HH*WW)            /* 12544 */
#define QKVC  (2*DQK + DV)          /* 608   */
#define ATTC  (KSZ*KSZ*GG)          /* 200   */
#define SCALE 0.17677669529663687f  /* (DV/G)^-0.5 = 32^-0.5 */

typedef float v2f __attribute__((ext_vector_type(2)));
typedef float v8f __attribute__((ext_vector_type(8)));

// ---------------------------------------------------------------------------
// fp32 WMMA GEMM, N-blocked:  C[M x N] = A[M x K] * W[N x K]^T (+ bias[N])
// One wave32 per 16x64 C tile: 4 x V_WMMA_F32_16X16X4_F32 per K-step sharing
// one A fragment (1 A-load + 4 B-loads per 4 WMMAs -> 1.25 loads/WMMA).
// A lane layout (ISA 7.12.2, 32-bit A 16x4): lane l -> row m=l%16,
//   VGPR0/1 = K = k0, k0+1 with k0 = (l>>4)*2.  B (4x16) mirrored.
// C/D layout: VGPR r -> C[mBase + r + (l>>4)*8][colBase + l%16].
// EXEC is all-1 around every WMMA; ragged N handled by clamping B loads
// (clamped columns never touch stored output columns) + masked stores.
// ---------------------------------------------------------------------------
__global__ __launch_bounds__(128)
void gemm_wmma_f32(const float* __restrict__ A, int lda,
                   const float* __restrict__ Wt, int ldw,
                   const float* __restrict__ bias,
                   float* __restrict__ C, int ldc,
                   int M, int N, int K, int nTiles64)
{
    const int wave = blockIdx.x * (blockDim.x >> 5) + (threadIdx.x >> 5);
    const int lane = threadIdx.x & 31;
    const int mTiles = M >> 4;
    if (wave >= mTiles * nTiles64) return;         // whole-wave uniform exit

    const int mt = wave / nTiles64;
    const int nt = wave - mt * nTiles64;
    const int mBase = mt << 4;
    const int nBase = nt << 6;                     // 64-wide N block

    const int row = mBase + (lane & 15);
    const int kl  = (lane >> 4) << 1;              // 0 or 2

    const float* __restrict__ ap = A + (size_t)row * lda + kl;

    int col[4];
    const float* bp[4];
    #pragma unroll
    for (int j = 0; j < 4; ++j) {
        col[j] = nBase + (j << 4) + (lane & 15);
        int cc = (col[j] < N) ? col[j] : (N - 1);  // clamp ragged columns
        bp[j]  = Wt + (size_t)cc * ldw + kl;
    }

    v8f acc0 = {0.f,0.f,0.f,0.f,0.f,0.f,0.f,0.f};
    v8f acc1 = acc0, acc2 = acc0, acc3 = acc0;

    #pragma unroll 2
    for (int k = 0; k < K; k += 4) {
        v2f a  = *(const v2f*)(ap + k);
        v2f b0 = *(const v2f*)(bp[0] + k);
        v2f b1 = *(const v2f*)(bp[1] + k);
        v2f b2 = *(const v2f*)(bp[2] + k);
        v2f b3 = *(const v2f*)(bp[3] + k);
        // (neg_a, A, neg_b, B, c_mod, C, reuse_a, reuse_b)
        acc0 = __builtin_amdgcn_wmma_f32_16x16x4_f32(false, a, false, b0, (short)0, acc0, false, false);
        acc1 = __builtin_amdgcn_wmma_f32_16x16x4_f32(false, a, false, b1, (short)0, acc1, false, false);
        acc2 = __builtin_amdgcn_wmma_f32_16x16x4_f32(false, a, false, b2, (short)0, acc2, false, false);
        acc3 = __builtin_amdgcn_wmma_f32_16x16x4_f32(false, a, false, b3, (short)0, acc3, false, false);
    }

    const int m0 = mBase + ((lane >> 4) << 3);
    v8f accs[4] = {acc0, acc1, acc2, acc3};
    #pragma unroll
    for (int j = 0; j < 4; ++j) {
        if (col[j] < N) {
            const float bv = bias ? bias[col[j]] : 0.f;
            #pragma unroll
            for (int r = 0; r < 8; ++r)
                C[(size_t)(m0 + r) * ldc + col[j]] = accs[j][r] + bv;
        }
    }
}

// ---------------------------------------------------------------------------
// quad = q * k * SCALE   (qkv pixel-major [p][608] -> quad [p][176])
// ---------------------------------------------------------------------------
__global__ __launch_bounds__(256)
void quad_kernel(const float* __restrict__ qkv, float* __restrict__ quad, int n)
{
    int i = blockIdx.x * blockDim.x + threadIdx.x;
    if (i >= n) return;
    int p = i / DQK, c = i - p * DQK;
    const float* r = qkv + (size_t)p * QKVC;
    quad[i] = r[c] * r[DQK + c] * SCALE;
}

// ---------------------------------------------------------------------------
// Grouped 5x5 conv (22 groups of 8ch -> 8ch) + bias + exact GELU.
// quad layout [b,h,w,176]; w_attn1 [oc][ic=8][5][5].
// ---------------------------------------------------------------------------
__global__ __launch_bounds__(256)
void conv_gelu_kernel(const float* __restrict__ quad,
                      const float* __restrict__ w1,
                      const float* __restrict__ b1,
                      float* __restrict__ out, int n)
{
    int i = blockIdx.x * blockDim.x + threadIdx.x;
    if (i >= n) return;
    int p  = i / DQK, oc = i - p * DQK;
    int b  = p / (HH * WW);
    int hw = p - b * HH * WW;
    int h  = hw / WW, w = hw - h * WW;
    int g  = oc >> 3;

    const float* __restrict__ wp = w1 + (size_t)oc * (8 * 25);
    float acc = b1[oc];
    for (int ki = 0; ki < KSZ; ++ki) {
        int hh = h + ki - 2;
        if ((unsigned)hh >= HH) continue;
        for (int kj = 0; kj < KSZ; ++kj) {
            int ww2 = w + kj - 2;
            if ((unsigned)ww2 >= WW) continue;
            const float* qp = quad + ((size_t)((b * HH + hh) * WW + ww2)) * DQK + g * 8;
            const float* wk = wp + ki * 5 + kj;
            #pragma unroll
            for (int ic = 0; ic < 8; ++ic)
                acc += qp[ic] * wk[ic * 25];
        }
    }
    out[i] = 0.5f * acc * (1.f + erff(acc * 0.70710678118654752f)); // exact GELU
}

// ---------------------------------------------------------------------------
// Softmax over the 25 taps per (pixel, group); attn layout [p][200] in-place.
// ---------------------------------------------------------------------------
__global__ __launch_bounds__(256)
void softmax_kernel(float* __restrict__ attn, int n)
{
    int i = blockIdx.x * blockDim.x + threadIdx.x;
    if (i >= n) return;
    int p = i / GG, g = i - p * GG;
    float* a = attn + (size_t)p * ATTC + g * 25;
    float mx = a[0];
    #pragma unroll
    for (int j = 1; j < 25; ++j) mx = fmaxf(mx, a[j]);
    float e[25], s = 0.f;
    #pragma unroll
    for (int j = 0; j < 25; ++j) { e[j] = expf(a[j] - mx); s += e[j]; }
    float inv = 1.f / s;
    #pragma unroll
    for (int j = 0; j < 25; ++j) a[j] = e[j] * inv;
}

// ---------------------------------------------------------------------------
// Ghost-head aggregation:
// out[p][c] = sum_idx vpad(b,c,h+ki-2,w+kj-2) * (ghm[c][idx]*attn + gha[c][idx])
// v lives in qkv[p][352+c]; zero-padded taps are skipped (contribute 0).
// ---------------------------------------------------------------------------
__global__ __launch_bounds__(256)
void agg_kernel(const float* __restrict__ qkv,
                const float* __restrict__ attn,
                const float* __restrict__ ghost,  // [2][256][5][5]
                float* __restrict__ out, int n)
{
    int i = blockIdx.x * blockDim.x + threadIdx.x;
    if (i >= n) return;
    int p  = i >> 8, c = i & 255;
    int b  = p / (HH * WW);
    int hw = p - b * HH * WW;
    int h  = hw / WW, w = hw - h * WW;
    int g  = c >> 5;

    const float* __restrict__ ghm = ghost + (size_t)c * 25;
    const float* __restrict__ gha = ghost + (size_t)DV * 25 + (size_t)c * 25;
    const float* __restrict__ ap  = attn + (size_t)p * ATTC + g * 25;

    float acc = 0.f;
    #pragma unroll
    for (int idx = 0; idx < 25; ++idx) {
        int ki = idx / 5, kj = idx - 5 * (idx / 5);
        int hh = h + ki - 2, ww2 = w + kj - 2;
        if ((unsigned)hh < HH && (unsigned)ww2 < WW) {
            float vv = qkv[((size_t)((b * HH + hh) * WW + ww2)) * QKVC + 2 * DQK + c];
            acc += vv * (ghm[idx] * ap[idx] + gha[idx]);
        }
    }
    out[i] = acc;
}

// ---------------------------------------------------------------------------
extern "C" void kernel_launch(void* const* d_in, const int* in_sizes, int n_in,
                              void* d_out, int out_size, void* d_ws, size_t ws_size,
                              hipStream_t stream)
{
    const float* x       = (const float*)d_in[0];
    const float* w_pre   = (const float*)d_in[1];
    const float* w_attn1 = (const float*)d_in[2];
    const float* b_attn1 = (const float*)d_in[3];
    const float* w_attn2 = (const float*)d_in[4];
    const float* b_attn2 = (const float*)d_in[5];
    const float* ghost   = (const float*)d_in[6];
    const float* w_post  = (const float*)d_in[7];
    const float* b_post  = (const float*)d_in[8];
    float* y = (float*)d_out;

    // Workspace layout (floats), all 16B-aligned offsets; total ~68 MB.
    float* ws     = (float*)d_ws;
    float* qkv    = ws;                                // 12544*608
    float* quad   = ws + (size_t)NPIX * QKVC;          // 12544*176
    float* aconv  = quad + (size_t)NPIX * DQK;         // 12544*176
    float* attn   = aconv + (size_t)NPIX * DQK;        // 12544*200
    float* outagg = attn + (size_t)NPIX * ATTC;        // 12544*256

    // 1) QKV projection: [12544 x 256] x [608 x 256]^T -> qkv
    {
        int nT = (QKVC + 63) / 64;                     // 10 (ragged: 608)
        int tiles = (NPIX / 16) * nT;
        gemm_wmma_f32<<<(tiles + 3) / 4, 128, 0, stream>>>(
            x, DIMC, w_pre, DIMC, (const float*)nullptr,
            qkv, QKVC, NPIX, QKVC, DIMC, nT);
    }
    // 2) quad = q*k*scale
    {
        int n = NPIX * DQK;
        quad_kernel<<<(n + 255) / 256, 256, 0, stream>>>(qkv, quad, n);
    }
    // 3) grouped conv 5x5 + bias + exact GELU
    {
        int n = NPIX * DQK;
        conv_gelu_kernel<<<(n + 255) / 256, 256, 0, stream>>>(quad, w_attn1, b_attn1, aconv, n);
    }
    // 4) attn logits: [12544 x 176] x [200 x 176]^T + b_attn2
    {
        int nT = (ATTC + 63) / 64;                     // 4 (ragged: 200)
        int tiles = (NPIX / 16) * nT;
        gemm_wmma_f32<<<(tiles + 3) / 4, 128, 0, stream>>>(
            aconv, DQK, w_attn2, DQK, b_attn2,
            attn, ATTC, NPIX, ATTC, DQK, nT);
    }
    // 5) softmax over 25 taps per (pixel, group)
    {
        int n = NPIX * GG;
        softmax_kernel<<<(n + 255) / 256, 256, 0, stream>>>(attn, n);
    }
    // 6) ghost-head aggregation -> outagg [p][256]
    {
        int n = NPIX * DV;
        agg_kernel<<<(n + 255) / 256, 256, 0, stream>>>(qkv, attn, ghost, outagg, n);
    }
    // 7) post projection: [12544 x 256] x [256 x 256]^T + b_post -> y (b,h,w,o)
    {
        int nT = DIMC / 64;                            // 4
        int tiles = (NPIX / 16) * nT;
        gemm_wmma_f32<<<(tiles + 3) / 4, 128, 0, stream>>>(
            outagg, DV, w_post, DV, b_post,
            y, DIMC, NPIX, DIMC, DV, nT);
    }
}